// MultiSegmentLoss_34943853920467
// MI455X (gfx1250) — compile-verified
//
#include <hip/hip_runtime.h>
#include <stdint.h>
#include <math.h>

// MultiSegmentLoss for MI455X (gfx1250, wave32).
// Memory-bound streaming reduction: 40 B/element * 3.2M elements = 128 MB -> ~5.5us @ 23.3 TB/s.
// Targets tile staged to LDS via CDNA5 async load; block reduction via V_WMMA_F32_16X16X4_F32
// (exact f32 row-sum against a ones matrix). Deterministic two-stage global reduction via d_ws.

#define TPB  256
#define MAXN 256

typedef float v2f __attribute__((ext_vector_type(2)));
typedef float v8f __attribute__((ext_vector_type(8)));

// accumulator slots: 0=loss_l 1=loss_c 2=loss_prop_l 3=loss_prop_c 4=loss_ct 5=Np 6=PN 7=pad
__global__ __launch_bounds__(TPB) void msl_main(
    const float* __restrict__ loc, const float* __restrict__ conf,
    const float* __restrict__ ploc, const float* __restrict__ pconf,
    const float* __restrict__ center, const float* __restrict__ priors,
    const float* __restrict__ targets, float* __restrict__ partials,
    int K, int N, int per_block)
{
    __shared__ float traw[3 * MAXN];
    __shared__ float ts0[MAXN], ts1[MAXN], tlab[MAXN];
    __shared__ float wsum[8][8];

    const int tid = threadIdx.x;
    const int b   = blockIdx.y;
    const int n3  = 3 * N;
    const float* tb = targets + (size_t)b * n3;

    // ---- Stage targets[b] (N x 3 floats) into LDS with CDNA5 async copy ----
    for (int i = tid; i < n3; i += TPB) {
        unsigned ldsa = (unsigned)(uintptr_t)(traw + i);   // low 32 bits of flat LDS ptr = LDS offset
        unsigned voff = (unsigned)(i * 4);
        asm volatile("global_load_async_to_lds_b32 %0, %1, %2"
                     :: "v"(ldsa), "v"(voff), "s"(tb) : "memory");
    }
#if __has_builtin(__builtin_amdgcn_s_wait_asynccnt)
    __builtin_amdgcn_s_wait_asynccnt(0);
#else
    asm volatile("s_wait_asynccnt 0" ::: "memory");
#endif
    __syncthreads();
    for (int j = tid; j < N; j += TPB) {
        ts0[j]  = traw[3 * j + 0];
        ts1[j]  = traw[3 * j + 1];
        tlab[j] = traw[3 * j + 2];
    }
    __syncthreads();

    float acc[8];
#pragma unroll
    for (int r = 0; r < 8; ++r) acc[r] = 0.0f;

    const float EPS = 1.1920928955078125e-07f;  // finfo(f32).eps
    const int kbase = blockIdx.x * per_block;
    int kend = kbase + per_block; if (kend > K) kend = K;

    for (int k = kbase + tid; k < kend; k += TPB) {
        const size_t idx = (size_t)b * K + k;
        const float  c   = priors[k];
        const float2 pre = ((const float2*)loc)[idx];
        const float pl = pre.x, pr = pre.y;

        // ---- match: argmin over N targets (first-min, matches jnp.argmin) ----
        float best_area = INFINITY; int best = 0;
        for (int j = 0; j < N; ++j) {
            float left  = (c - ts0[j]) * 256.0f;
            float right = (ts1[j] - c) * 256.0f;
            float area  = (left < 0.0f || right < 0.0f) ? 512.0f : (left + right);
            if (area < best_area) { best_area = area; best = j; }
        }
        const float tl = (c - ts0[best]) * 256.0f;
        const float tr = (ts1[best] - c) * 256.0f;
        const int conft = (best_area >= 512.0f) ? 0 : (int)tlab[best];

        const float inter = fminf(pl, tl) + fminf(pr, tr);
        const float uni   = pl + pr + (tl + tr) - inter;
        const float iou   = inter / fmaxf(uni, EPS);
        const int pconft  = (iou < 0.5f) ? 0 : conft;

        const float2 pld    = ((const float2*)ploc)[idx];
        const float  prop_w = pl + pr;

        if (conft > 0) {
            // GIoU loss
            float ac   = fmaxf(pl, tl) + fmaxf(pr, tr);
            float giou = iou - (ac - uni) / fmaxf(ac, EPS);
            acc[0] += 1.0f - giou;
            // centerness BCE against clipped IoU of refined location
            float cl0  = 0.5f * prop_w * pld.x + pl;
            float cl1  = 0.5f * prop_w * pld.y + pr;
            float in2  = fminf(cl0, tl) + fminf(cl1, tr);
            float un2  = cl0 + cl1 + (tl + tr) - in2;
            float iou2 = fmaxf(in2 / fmaxf(un2, EPS), 0.0f);
            float lg   = center[idx];
            acc[4] += fmaxf(lg, 0.0f) - lg * iou2 + log1pf(expf(-fabsf(lg)));
            acc[5] += 1.0f;  // Np
        }
        if (pconft > 0) {
            float d = 0.5f * prop_w;
            acc[2] += fabsf(pld.x - (tl - pl) / d) + fabsf(pld.y - (tr - pr) / d);
            acc[6] += 1.0f;  // PN
        }
        {   // focal loss on conf (all elements)
            const float2 cd = ((const float2*)conf)[idx];
            float mx = fmaxf(cd.x, cd.y);
            float e0 = expf(cd.x - mx), e1 = expf(cd.y - mx);
            float pt = ((conft == 0) ? e0 : e1) / (e0 + e1) + 1e-6f;
            float al = (conft == 0) ? 0.25f : 0.75f;
            float om = 1.0f - pt;
            acc[1] -= om * om * al * logf(pt);
        }
        {   // focal loss on prop_conf (all elements)
            const float2 cd = ((const float2*)pconf)[idx];
            float mx = fmaxf(cd.x, cd.y);
            float e0 = expf(cd.x - mx), e1 = expf(cd.y - mx);
            float pt = ((pconft == 0) ? e0 : e1) / (e0 + e1) + 1e-6f;
            float al = (pconft == 0) ? 0.25f : 0.75f;
            float om = 1.0f - pt;
            acc[3] -= om * om * al * logf(pt);
        }
    }

    // ---- wave32 tree reduce (deterministic) ----
#pragma unroll
    for (int o = 16; o > 0; o >>= 1)
#pragma unroll
        for (int r = 0; r < 8; ++r)
            acc[r] += __shfl_down(acc[r], (unsigned)o, 32);

    const int lane = tid & 31, wid = tid >> 5;
    if (lane == 0)
        for (int r = 0; r < 8; ++r) wsum[wid][r] = acc[r];
    __syncthreads();

    // ---- block reduce 8 waves x 8 accs with WMMA f32 16x16x4: D = A * ones + C ----
    if (tid < 32) {   // wave 0, full EXEC (WMMA requires all-ones EXEC)
        const int  m  = tid & 15;
        const bool hi = tid >= 16;
        v2f a1 = {0.0f, 0.0f}, a2 = {0.0f, 0.0f};
        if (m < 8) {   // A[m,k] = wave-k partial of accumulator m; rows 8..15 zero
            a1.x = hi ? wsum[2][m] : wsum[0][m];   // lanes 0-15: K=0,1 ; lanes 16-31: K=2,3
            a1.y = hi ? wsum[3][m] : wsum[1][m];
            a2.x = hi ? wsum[6][m] : wsum[4][m];
            a2.y = hi ? wsum[7][m] : wsum[5][m];
        }
        const size_t blin = (size_t)blockIdx.y * gridDim.x + blockIdx.x;
#if __has_builtin(__builtin_amdgcn_wmma_f32_16x16x4_f32)
        v2f ones = {1.0f, 1.0f};
        v8f cc = {};
        cc = __builtin_amdgcn_wmma_f32_16x16x4_f32(false, a1, false, ones, (short)0, cc, false, false);
        cc = __builtin_amdgcn_wmma_f32_16x16x4_f32(false, a2, false, ones, (short)0, cc, false, false);
        if (tid == 0) {   // lane 0 holds column N=0: D[r,0] in cc[r]
#pragma unroll
            for (int r = 0; r < 8; ++r) partials[blin * 8 + r] = cc[r];
        }
#else
        if (tid == 0) {
            for (int r = 0; r < 8; ++r) {
                float v = 0.0f;
                for (int w = 0; w < 8; ++w) v += wsum[w][r];
                partials[blin * 8 + r] = v;
            }
        }
#endif
    }
}

__global__ void msl_finalize(const float* __restrict__ partials,
                             float* __restrict__ out, int nblk)
{
    __shared__ float s[8];
    const int t = threadIdx.x;
    if (t < 8) {
        float v = 0.0f;
        for (int i = 0; i < nblk; ++i) v += partials[(size_t)i * 8 + t];  // fixed order: deterministic
        s[t] = v;
    }
    __syncthreads();
    if (t == 0) {
        float Np = fmaxf(s[5], 1.0f);
        float PN = fmaxf(s[6], 1.0f);
        out[0] = s[0] / Np;   // loss_l / Np
        out[1] = s[1] / Np;   // loss_c / Np
        out[2] = s[2] / PN;   // loss_prop_l / PN
        out[3] = s[3] / PN;   // loss_prop_c / PN
        out[4] = s[4] / Np;   // loss_ct / Np
    }
}

extern "C" void kernel_launch(void* const* d_in, const int* in_sizes, int n_in,
                              void* d_out, int out_size, void* d_ws, size_t ws_size,
                              hipStream_t stream) {
    (void)n_in; (void)out_size;
    const float* loc     = (const float*)d_in[0];
    const float* conf    = (const float*)d_in[1];
    const float* ploc    = (const float*)d_in[2];
    const float* pconf   = (const float*)d_in[3];
    const float* center  = (const float*)d_in[4];
    const float* priors  = (const float*)d_in[5];
    const float* targets = (const float*)d_in[6];

    const long K = in_sizes[5];                 // priors: (K,1)
    const long B = in_sizes[0] / (2 * K);       // loc_data: (B,K,2)
    long N = in_sizes[6] / (3 * B);             // targets: (B,N,3)
    if (N > MAXN) N = MAXN;

    int per_block = 2048;                       // 8 k-iters per thread; amortizes target staging
    long kchunks = (K + per_block - 1) / per_block;
    while ((size_t)(B * kchunks * 8 * sizeof(float)) > ws_size && (long)per_block < K) {
        per_block <<= 1;
        kchunks = (K + per_block - 1) / per_block;
    }

    dim3 grid((unsigned)kchunks, (unsigned)B);
    msl_main<<<grid, TPB, 0, stream>>>(loc, conf, ploc, pconf, center, priors, targets,
                                       (float*)d_ws, (int)K, (int)N, per_block);
    msl_finalize<<<1, 64, 0, stream>>>((const float*)d_ws, (float*)d_out,
                                       (int)(kchunks * B));
}